// GraphAttention_24979529793774
// MI455X (gfx1250) — compile-verified
//
#include <hip/hip_runtime.h>

#define Hh   1024
#define Bb   64
#define Nn   512
#define NEGC 1e30f

typedef float v2f __attribute__((ext_vector_type(2)));
typedef float v8f __attribute__((ext_vector_type(8)));

// ---------------------------------------------------------------------------
// Kernel 1: k_score[b*N+n] = dot(K[b,n,:], Wk)   (Wk = W_att[H:2H])
// One wave32 per row; float4 coalesced loads; shfl-xor reduction.
// ---------------------------------------------------------------------------
__global__ __launch_bounds__(256) void kscore_kernel(const float* __restrict__ K,
                                                     const float* __restrict__ Watt,
                                                     float* __restrict__ kscore) {
    const int wave = threadIdx.x >> 5;
    const int lane = threadIdx.x & 31;
    const int row  = blockIdx.x * 8 + wave;            // [0, B*N)
    const float* __restrict__ Kr = K + (size_t)row * Hh;
    const float* __restrict__ Wk = Watt + Hh;
    float acc = 0.f;
#pragma unroll
    for (int i = 0; i < 8; ++i) {
        const int idx = lane * 4 + i * 128;
        const float4 kv = *(const float4*)(Kr + idx);
        const float4 wv = *(const float4*)(Wk + idx);
        acc += kv.x * wv.x + kv.y * wv.y + kv.z * wv.z + kv.w * wv.w;
    }
#pragma unroll
    for (int off = 16; off >= 1; off >>= 1)
        acc += __shfl_xor(acc, off, 32);
    if (lane == 0) kscore[row] = acc;
}

// ---------------------------------------------------------------------------
// Kernel 2: per-batch masked softmax. Also computes q_score = dot(Q[b], Wq)
// and adds b_att. Writes attn_weight (B,N) straight into d_out.
// ---------------------------------------------------------------------------
__global__ __launch_bounds__(256) void softmax_kernel(const float* __restrict__ Q,
                                                      const float* __restrict__ Watt,
                                                      const float* __restrict__ batt,
                                                      const int*   __restrict__ adj,
                                                      const float* __restrict__ kscore,
                                                      float* __restrict__ w_out) {
    const int b = blockIdx.x;
    const int t = threadIdx.x;
    __shared__ float red[256];

    // q_score partial dot
    float qs = 0.f;
#pragma unroll
    for (int i = 0; i < 4; ++i) {
        const int idx = t + i * 256;
        qs += Q[b * Hh + idx] * Watt[idx];
    }
    red[t] = qs;
    __syncthreads();
    for (int s = 128; s > 0; s >>= 1) {
        if (t < s) red[t] += red[t + s];
        __syncthreads();
    }
    const float base = red[0] + batt[0];
    __syncthreads();

    const int n0 = t, n1 = t + 256;
    const float a0 = kscore[b * Nn + n0] + base - (1.f - (float)adj[b * Nn + n0]) * NEGC;
    const float a1 = kscore[b * Nn + n1] + base - (1.f - (float)adj[b * Nn + n1]) * NEGC;

    red[t] = fmaxf(a0, a1);
    __syncthreads();
    for (int s = 128; s > 0; s >>= 1) {
        if (t < s) red[t] = fmaxf(red[t], red[t + s]);
        __syncthreads();
    }
    const float M = red[0];
    __syncthreads();

    const float e0 = expf(a0 - M), e1 = expf(a1 - M);
    red[t] = e0 + e1;
    __syncthreads();
    for (int s = 128; s > 0; s >>= 1) {
        if (t < s) red[t] += red[t + s];
        __syncthreads();
    }
    const float inv = 1.f / red[0];
    w_out[b * Nn + n0] = e0 * inv;
    w_out[b * Nn + n1] = e1 * inv;
}

// ---------------------------------------------------------------------------
// Kernel 3: u0[b,k] = sum_n w*s*V[b,n,k] ; u1[b,k] = sum_n w*o*V[b,n,k]
// Block per b; 256 threads x float4 cover all 1024 k. V read exactly once.
// ---------------------------------------------------------------------------
__global__ __launch_bounds__(256) void vreduce_kernel(const float* __restrict__ V,
                                                      const float* __restrict__ w,
                                                      const int*   __restrict__ smask,
                                                      const int*   __restrict__ omask,
                                                      float* __restrict__ u0,
                                                      float* __restrict__ u1) {
    const int b = blockIdx.x;
    const int t = threadIdx.x;
    __shared__ float ws0[Nn];
    __shared__ float ws1[Nn];
#pragma unroll
    for (int i = 0; i < 2; ++i) {
        const int n = t + i * 256;
        const float wv = w[b * Nn + n];
        ws0[n] = wv * (float)smask[b * Nn + n];
        ws1[n] = wv * (float)omask[b * Nn + n];
    }
    __syncthreads();

    const int k0 = t * 4;
    const float* __restrict__ Vb = V + (size_t)b * Nn * Hh + k0;
    float4 a0 = make_float4(0.f, 0.f, 0.f, 0.f);
    float4 a1 = make_float4(0.f, 0.f, 0.f, 0.f);
#pragma unroll 4
    for (int n = 0; n < Nn; ++n) {
        const float4 v = *(const float4*)(Vb + (size_t)n * Hh);
        const float c0 = ws0[n];
        const float c1 = ws1[n];
        a0.x += c0 * v.x; a0.y += c0 * v.y; a0.z += c0 * v.z; a0.w += c0 * v.w;
        a1.x += c1 * v.x; a1.y += c1 * v.y; a1.z += c1 * v.z; a1.w += c1 * v.w;
    }
    *(float4*)(u0 + b * Hh + k0) = a0;
    *(float4*)(u1 + b * Hh + k0) = a1;
}

// ---------------------------------------------------------------------------
// Kernel 4: out(64x1024) = u0 @ Wr0^T + u1 @ Wr1^T via V_WMMA_F32_16X16X4_F32.
// One wave per 16x16 output tile; K=1024 in steps of 4; accumulator chained
// across both weight matrices. A/B fragment layout per ISA 7.12.2 (32-bit
// 16x4): lanes 0-15 hold K=k0,k0+1; lanes 16-31 hold K=k0+2,k0+3 (same M/N).
// ---------------------------------------------------------------------------
__global__ __launch_bounds__(32) void wmma_gemm_kernel(const float* __restrict__ u0,
                                                       const float* __restrict__ u1,
                                                       const float* __restrict__ Wr0,
                                                       const float* __restrict__ Wr1,
                                                       float* __restrict__ out) {
    const int lane = threadIdx.x;          // 0..31
    const int n0   = blockIdx.x * 16;      // H tile
    const int m0   = blockIdx.y * 16;      // B tile
    const int lh   = lane & 15;
    const int hi   = lane >> 4;

    const float* __restrict__ A0 = u0  + (size_t)(m0 + lh) * Hh;
    const float* __restrict__ A1 = u1  + (size_t)(m0 + lh) * Hh;
    const float* __restrict__ B0 = Wr0 + (size_t)(n0 + lh) * Hh;   // B[k][n] = Wr[n][k]
    const float* __restrict__ B1 = Wr1 + (size_t)(n0 + lh) * Hh;

    v8f acc = {0.f, 0.f, 0.f, 0.f, 0.f, 0.f, 0.f, 0.f};

    for (int k0 = 0; k0 < Hh; k0 += 4) {
        const int kk = k0 + hi * 2;
        const v2f a0 = *(const v2f*)(A0 + kk);
        const v2f b0 = *(const v2f*)(B0 + kk);
        acc = __builtin_amdgcn_wmma_f32_16x16x4_f32(false, a0, false, b0,
                                                    (short)0, acc, false, false);
        const v2f a1 = *(const v2f*)(A1 + kk);
        const v2f b1 = *(const v2f*)(B1 + kk);
        acc = __builtin_amdgcn_wmma_f32_16x16x4_f32(false, a1, false, b1,
                                                    (short)0, acc, false, false);
    }

    // C/D layout: VGPR g -> rows m0+g (lanes 0-15) and m0+g+8 (lanes 16-31)
#pragma unroll
    for (int g = 0; g < 8; ++g) {
        out[(size_t)(m0 + g + 8 * hi) * Hh + n0 + lh] = acc[g];
    }
}

// ---------------------------------------------------------------------------
extern "C" void kernel_launch(void* const* d_in, const int* in_sizes, int n_in,
                              void* d_out, int out_size, void* d_ws, size_t ws_size,
                              hipStream_t stream) {
    const float* Q     = (const float*)d_in[0];
    const float* K     = (const float*)d_in[1];
    const float* V     = (const float*)d_in[2];
    const int*   adj   = (const int*)d_in[3];
    const int*   smask = (const int*)d_in[4];
    const int*   omask = (const int*)d_in[5];
    const float* Watt  = (const float*)d_in[6];
    const float* batt  = (const float*)d_in[7];
    const float* Wr0   = (const float*)d_in[8];
    const float* Wr1   = (const float*)d_in[9];

    float* w_out    = (float*)d_out;             // attn_weight: B*N floats
    float* attn_out = (float*)d_out + Bb * Nn;   // attn_sum:    B*H floats

    float* kscore = (float*)d_ws;                // B*N
    float* u0     = kscore + Bb * Nn;            // B*H
    float* u1     = u0 + Bb * Hh;                // B*H

    kscore_kernel <<<Bb * Nn / 8, 256, 0, stream>>>(K, Watt, kscore);
    softmax_kernel<<<Bb,          256, 0, stream>>>(Q, Watt, batt, adj, kscore, w_out);
    vreduce_kernel<<<Bb,          256, 0, stream>>>(V, w_out, smask, omask, u0, u1);
    wmma_gemm_kernel<<<dim3(Hh / 16, Bb / 16), 32, 0, stream>>>(u0, u1, Wr0, Wr1, attn_out);
}